// FeedForwardTree_9191230013859
// MI455X (gfx1250) — compile-verified
//
#include <hip/hip_runtime.h>
#include <stdint.h>

// FeedForwardTree on MI455X (gfx1250, wave32).
// Latency/memory-bound dependent-gather workload: no matmul structure, so no
// WMMA; instead we use the CDNA5 async global->LDS path (ASYNCcnt) to stage
// the next tree node while finishing the current level, wave32 butterfly
// shuffles for the 128-wide logsumexp, and coalesced b128 loads.

#define DEPTH     10
#define B_Q       64
#define IN_DIM    128
#define OUT_DIM   256
#define NPAIR     (B_Q * OUT_DIM)          // 16384 tree walks
#define NODES     ((1 << DEPTH) - 1)       // 1023 nodes per output tree
#define ROW_F     (2 * IN_DIM)             // 256 floats (1KB) per node row
#define WAVES_PER_BLOCK 8

__device__ __forceinline__ float fast_logaddexp(float x, float y) {
    float m = fmaxf(x, y);
    float d = fabsf(x - y);
    return m + __logf(1.0f + __expf(-d));
}

// Deterministic per-(depth,pair) uniform in [0,1). (JAX threefry key-split
// chain is not bit-reproducible here; any fixed counter hash is deterministic
// across graph replays, which is what the harness requires.)
__device__ __forceinline__ float uniform01(uint32_t d, uint32_t n) {
    uint32_t x = 0x2545F491u ^ (n * 0x9E3779B9u) ^ (d * 0x85EBCA6Bu);
    x ^= x >> 16; x *= 0x7FEB352Du;
    x ^= x >> 15; x *= 0x846CA68Bu;
    x ^= x >> 16;
    return (float)(x >> 8) * 0x1.0p-24f;
}

// CDNA5 async tensor path: 16B global -> LDS, tracked by ASYNCcnt.
// VDST supplies the per-lane LDS byte address (CDNA5 ISA).
__device__ __forceinline__ void async_copy_b128(uint32_t lds_off, const float* gaddr) {
    asm volatile("global_load_async_to_lds_b128 %0, %1, off"
                 :: "v"(lds_off), "v"(gaddr) : "memory");
}
__device__ __forceinline__ void wait_async0() {
    asm volatile("s_wait_asynccnt 0" ::: "memory");
}
__device__ __forceinline__ void wait_ds0() {
    asm volatile("s_wait_dscnt 0" ::: "memory");
}

__global__ void __launch_bounds__(256)
tree_walk_kernel(const float* __restrict__ query,     // (64,128)
                 const float* __restrict__ key_p,     // (256,1023,2,128)
                 const float* __restrict__ value_p,   // (256,1024,1)
                 float* __restrict__ out_val,         // (64,256) flat
                 float* __restrict__ support_ws) {    // (64*256) scratch
    __shared__ float kbuf[WAVES_PER_BLOCK * ROW_F];   // 1KB node row per wave
    const int wv   = threadIdx.x >> 5;
    const int lane = threadIdx.x & 31;
    const int n    = blockIdx.x * WAVES_PER_BLOCK + wv;   // pair id
    const int o    = n & (OUT_DIM - 1);
    const int bq   = n >> 8;

    float* kwv = &kbuf[wv * ROW_F];
    // Low 32 bits of a flat pointer into the LDS aperture == LDS byte offset.
    const uint32_t ldsbase = (uint32_t)(uintptr_t)kwv;

    const float* keybase = key_p + (size_t)o * NODES * ROW_F;

    // Lane handles dims [4*lane, 4*lane+4): fully coalesced b128 per wave.
    float4 qv = *(const float4*)(query + bq * IN_DIM + lane * 4);

    // Depth 0: node 0 straight from global.
    float4 k1v = *(const float4*)(keybase + lane * 4);
    float4 k2v = *(const float4*)(keybase + IN_DIM + lane * 4);

    int   ix = 0;
    float support = 0.0f;

    #pragma unroll
    for (int d = 0; d < DEPTH; ++d) {
        const float q[4]  = {qv.x, qv.y, qv.z, qv.w};
        const float k1[4] = {k1v.x, k1v.y, k1v.z, k1v.w};
        const float k2[4] = {k2v.x, k2v.y, k2v.z, k2v.w};
        float neg[4];
        float lm = -3.4e38f;
        #pragma unroll
        for (int i = 0; i < 4; ++i) {
            float a1 = -fast_logaddexp(-k1[i], -q[i]);
            float a2 = -fast_logaddexp(-k2[i],  q[i]);
            float s  = fast_logaddexp(a1, a2);
            neg[i] = -s;                         // for -logsumexp(-s)
            lm = fmaxf(lm, neg[i]);
        }
        // wave32 butterfly max (all lanes end with the wave max)
        #pragma unroll
        for (int off = 16; off > 0; off >>= 1)
            lm = fmaxf(lm, __shfl_xor(lm, off, 32));
        float psum = 0.0f;
        #pragma unroll
        for (int i = 0; i < 4; ++i) psum += __expf(neg[i] - lm);
        #pragma unroll
        for (int off = 16; off > 0; off >>= 1)
            psum += __shfl_xor(psum, off, 32);

        const float s_sc = -(lm + __logf(psum));         // -logsumexp(-s)
        const float p    = 1.0f / (1.0f + __expf(-s_sc)); // sigmoid
        const int   bit  = (uniform01((uint32_t)d, (uint32_t)n) < p) ? 1 : 0;
        ix = 2 * ix + bit;

        // The next node is now fully determined: stage its 1KB row into LDS
        // asynchronously while we finish the support update.
        if (d < DEPTH - 1) {
            const float* nrow =
                keybase + (size_t)((2u << d) - 1u + (uint32_t)ix) * ROW_F;
            wait_ds0();  // prior ds_loads of kwv fully drained before overwrite
            const uint32_t loff = ldsbase + (uint32_t)lane * 16u;
            async_copy_b128(loff,        nrow + lane * 4);            // k1 half
            async_copy_b128(loff + 512u, nrow + IN_DIM + lane * 4);   // k2 half
        }

        const float ss = bit ? s_sc : -s_sc;
        support = (d == 0) ? ss : -fast_logaddexp(-support, -ss);

        if (d < DEPTH - 1) {
            wait_async0();
            k1v = *(const float4*)(kwv + lane * 4);
            k2v = *(const float4*)(kwv + IN_DIM + lane * 4);
        }
    }

    if (lane == 0) {
        out_val[n]    = value_p[(size_t)o * (1 << DEPTH) + ix];
        support_ws[n] = support;
    }
}

// logsumexp over o (256 values) per query row b -> d_out[0:64]
__global__ void __launch_bounds__(256)
support_reduce_kernel(const float* __restrict__ ws, float* __restrict__ out) {
    __shared__ float sm[8];
    const int b = blockIdx.x;
    const int t = threadIdx.x;
    const float v = ws[b * OUT_DIM + t];

    float m = v;
    #pragma unroll
    for (int off = 16; off > 0; off >>= 1)
        m = fmaxf(m, __shfl_xor(m, off, 32));
    if ((t & 31) == 0) sm[t >> 5] = m;
    __syncthreads();
    float M = sm[0];
    #pragma unroll
    for (int i = 1; i < 8; ++i) M = fmaxf(M, sm[i]);
    __syncthreads();

    float e = __expf(v - M);
    #pragma unroll
    for (int off = 16; off > 0; off >>= 1)
        e += __shfl_xor(e, off, 32);
    if ((t & 31) == 0) sm[t >> 5] = e;
    __syncthreads();
    if (t == 0) {
        float ssum = 0.0f;
        #pragma unroll
        for (int i = 0; i < 8; ++i) ssum += sm[i];
        out[b] = M + __logf(ssum);
    }
}

extern "C" void kernel_launch(void* const* d_in, const int* in_sizes, int n_in,
                              void* d_out, int out_size, void* d_ws, size_t ws_size,
                              hipStream_t stream) {
    (void)in_sizes; (void)n_in; (void)out_size; (void)ws_size;
    const float* query   = (const float*)d_in[0];   // 64*128
    const float* key_p   = (const float*)d_in[1];   // 256*1023*2*128
    const float* value_p = (const float*)d_in[2];   // 256*1024
    float* out = (float*)d_out;                     // [0:64) support, [64:) val
    float* ws  = (float*)d_ws;                      // 16384 floats

    tree_walk_kernel<<<NPAIR / WAVES_PER_BLOCK, 256, 0, stream>>>(
        query, key_p, value_p, out + B_Q, ws);
    support_reduce_kernel<<<B_Q, 256, 0, stream>>>(ws, out);
}